// LCAModel_27771258536510
// MI455X (gfx1250) — compile-verified
//
#include <hip/hip_runtime.h>
#include <stdint.h>

// ---------------- LCA hyperparameters (from reference) ----------------
#define NUM_SIMS   2048
#define NUM_DIM    8
#define NUM_STEPS  1500
#define THRESHOLD  1.0f
#define LEAK       0.1f
#define COMPETITION 0.1f
#define DT         0.01f
#define SQRT_STEP  0.0316227766016838f   // sqrt(0.001)

#define T_OUT      (NUM_STEPS + 1)
#define ROW_ELEMS  (NUM_SIMS * NUM_DIM)          // 16384 floats per time-row of noise

// scan tiling: one wave32 workgroup owns 32 sims; noise double-buffered by TDM
#define SIMS_PER_BLOCK 32
#define TILE_COLS  (SIMS_PER_BLOCK * NUM_DIM)    // 256 floats per block per step
#define T_TILE     25
#define N_TILES    (NUM_STEPS / T_TILE)          // 60 * 25 = 1500

#if defined(__has_builtin)
#if __has_builtin(__builtin_amdgcn_tensor_load_to_lds) && __has_builtin(__builtin_amdgcn_s_wait_tensorcnt)
#define HAVE_TDM 1
#endif
#endif
#ifndef HAVE_TDM
#define HAVE_TDM 0
#endif

// ---------------- TDM descriptor helpers (CDNA5 D#, 2D tile) ----------------
#if HAVE_TDM
typedef unsigned int v4u __attribute__((ext_vector_type(4)));
typedef int          v8i __attribute__((ext_vector_type(8)));
typedef int          v4i __attribute__((ext_vector_type(4)));

__device__ __forceinline__ void tdm_load_tile(uint64_t gaddr_bytes, unsigned lds_byte_addr) {
  // D# group 0: count=1, lds_addr, 57-bit global addr, type=2 ("image")
  v4u g0;
  g0[0] = 1u;
  g0[1] = lds_byte_addr;
  g0[2] = (unsigned)(gaddr_bytes & 0xFFFFFFFFull);
  g0[3] = (unsigned)((gaddr_bytes >> 32) & 0x01FFFFFFull) | (2u << 30);
  // D# group 1: data_size=4B (code 2), tile 256 x 25, tensor_dim0_stride = 16384
  const unsigned TD0 = (unsigned)ROW_ELEMS;   // tensor_dim0 (elements)
  const unsigned TD1 = 1504u;                 // tensor_dim1 (>= NUM_STEPS)
  v8i g1;
  g1[0] = (int)(2u << 16);                                    // wg_mask=0, data_size=2 (4B)
  g1[1] = (int)((TD0 & 0xFFFFu) << 16);                       // bits63:48 tensor_dim0 lo16
  g1[2] = (int)((TD0 >> 16) | ((TD1 & 0xFFFFu) << 16));       // tdim0 hi16 | tdim1 lo16
  g1[3] = (int)((TD1 >> 16) | ((unsigned)TILE_COLS << 16));   // tdim1 hi16 | tile_dim0=256
  g1[4] = (int)T_TILE;                                        // tile_dim1=25, tile_dim2=0
  g1[5] = (int)(unsigned)ROW_ELEMS;                           // tensor_dim0_stride lo32
  g1[6] = 0;                                                  // stride hi16 | dim1_stride lo16
  g1[7] = 0;
  v4i gz4 = {0, 0, 0, 0};                 // groups 2/3 unused (2D tensor)
  v8i gz8 = {0, 0, 0, 0, 0, 0, 0, 0};     // trailing group (clang-23 6-arg form)
  __builtin_amdgcn_tensor_load_to_lds(g0, g1, gz4, gz4, gz8, 0);
}
#endif

// Low 32 bits of a generic pointer to __shared__ == LDS byte address (aperture mapping).
__device__ __forceinline__ unsigned lds_addr_of(const void* p) {
  return (unsigned)(uintptr_t)p;
}

// ---------------- Kernel B: 1500-step serial scan, 1 wave32 per 32 sims ----------------
// (defined first so the disasm snippet shows the TDM issue + s_wait_tensorcnt sequence)
__global__ void __launch_bounds__(SIMS_PER_BLOCK)
lca_scan(const float* __restrict__ input, const float* __restrict__ noise,
         float* __restrict__ out) {
#if HAVE_TDM
  __shared__ __align__(16) float nbuf[2][T_TILE * TILE_COLS];   // 2 x 25.6 KB
#endif
  const int tid = threadIdx.x;
  const int b = blockIdx.x * SIMS_PER_BLOCK + tid;

  float pre[NUM_DIM], act[NUM_DIM], inp[NUM_DIM];
#pragma unroll
  for (int d = 0; d < NUM_DIM; d++) {
    pre[d] = 0.0f; act[d] = 0.0f;
    inp[d] = input[b * NUM_DIM + d];
  }

  float* out_pre = out;
  float* out_act = out + (size_t)NUM_SIMS * NUM_DIM * T_OUT;

  // t = 0 initial zeros (d_out is poisoned, must be written)
#pragma unroll
  for (int d = 0; d < NUM_DIM; d++) {
    size_t base = ((size_t)b * NUM_DIM + d) * T_OUT;
    out_pre[base] = 0.0f;
    out_act[base] = 0.0f;
  }

#if HAVE_TDM
  const uint64_t nbase = (uint64_t)(uintptr_t)noise +
                         (uint64_t)blockIdx.x * TILE_COLS * sizeof(float);
  const uint64_t tile_bytes = (uint64_t)T_TILE * ROW_ELEMS * sizeof(float);
  // prime the pipeline: tile 0 -> buffer 0
  tdm_load_tile(nbase, lds_addr_of(&nbuf[0][0]));
#endif

  for (int tile = 0; tile < N_TILES; ++tile) {
    const int cur = tile & 1;
#if HAVE_TDM
    if (tile + 1 < N_TILES) {
      // prefetch next tile into the other buffer, then release current tile:
      // TENSORcnt is in-order per wave -> waiting for <=1 completes tile `tile`.
      tdm_load_tile(nbase + (uint64_t)(tile + 1) * tile_bytes,
                    lds_addr_of(&nbuf[cur ^ 1][0]));
      __builtin_amdgcn_s_wait_tensorcnt(1);
    } else {
      __builtin_amdgcn_s_wait_tensorcnt(0);
    }
    asm volatile("" ::: "memory");   // keep LDS reads below the tensorcnt wait
#endif

#pragma unroll 1
    for (int s = 0; s < T_TILE; ++s) {
      const int t = tile * T_TILE + s;
#if HAVE_TDM
      const float* nz = &nbuf[cur][s * TILE_COLS + tid * NUM_DIM];
#else
      const float* nz = noise + (size_t)t * ROW_ELEMS + (size_t)b * NUM_DIM;
#endif
      // active = all(|act| < 1);  rec = sum(act @ gamma) == -0.7 * sum(act)
      float S = 0.0f, mx = 0.0f;
#pragma unroll
      for (int d = 0; d < NUM_DIM; d++) {
        S += act[d];
        mx = fmaxf(mx, fabsf(act[d]));
      }
      const float active = (mx < THRESHOLD) ? 1.0f : 0.0f;
      const float rec = -(COMPETITION * (float)(NUM_DIM - 1)) * S;
      const float gdt = active * DT;
      const float gnz = active * SQRT_STEP;   // noise_std = 1

#pragma unroll
      for (int d = 0; d < NUM_DIM; d++) {
        float p = pre[d];
        p = fmaf((inp[d] - LEAK * p) + rec, gdt, p);
        p = fmaf(gnz, nz[d], p);
        pre[d] = p;
        const float a = fmaxf(p, 0.0f);
        act[d] = a;
        const size_t base = ((size_t)b * NUM_DIM + d) * T_OUT + (size_t)(t + 1);
        out_pre[base] = p;   // consecutive t hit the same 128B line -> L2 absorbs
        out_act[base] = a;
      }
    }
  }
}

// ---------------- Threefry-2x32-20 (JAX PRNG) ----------------
__device__ __forceinline__ uint32_t rotl32(uint32_t x, int r) {
  return (x << r) | (x >> (32 - r));
}

__device__ __forceinline__ void threefry2x32(uint32_t k0, uint32_t k1,
                                             uint32_t x0, uint32_t x1,
                                             uint32_t* o0, uint32_t* o1) {
  const uint32_t k2 = k0 ^ k1 ^ 0x1BD11BDAu;
  const int rotA[4] = {13, 15, 26, 6};
  const int rotB[4] = {17, 29, 16, 24};
  x0 += k0; x1 += k1;
#pragma unroll
  for (int i = 0; i < 4; i++) { x0 += x1; x1 = rotl32(x1, rotA[i]); x1 ^= x0; }
  x0 += k1; x1 += k2 + 1u;
#pragma unroll
  for (int i = 0; i < 4; i++) { x0 += x1; x1 = rotl32(x1, rotB[i]); x1 ^= x0; }
  x0 += k2; x1 += k0 + 2u;
#pragma unroll
  for (int i = 0; i < 4; i++) { x0 += x1; x1 = rotl32(x1, rotA[i]); x1 ^= x0; }
  x0 += k0; x1 += k1 + 3u;
#pragma unroll
  for (int i = 0; i < 4; i++) { x0 += x1; x1 = rotl32(x1, rotB[i]); x1 ^= x0; }
  x0 += k1; x1 += k2 + 4u;
#pragma unroll
  for (int i = 0; i < 4; i++) { x0 += x1; x1 = rotl32(x1, rotA[i]); x1 ^= x0; }
  x0 += k2; x1 += k0 + 5u;
  *o0 = x0; *o1 = x1;
}

// Giles single-precision erfinv
__device__ __forceinline__ float erfinv_f(float x) {
  float w = -__logf((1.0f - x) * (1.0f + x));
  float p;
  if (w < 5.0f) {
    w -= 2.5f;
    p = 2.81022636e-08f;
    p = fmaf(p, w, 3.43273939e-07f);
    p = fmaf(p, w, -3.5233877e-06f);
    p = fmaf(p, w, -4.39150654e-06f);
    p = fmaf(p, w, 0.00021858087f);
    p = fmaf(p, w, -0.00125372503f);
    p = fmaf(p, w, -0.00417768164f);
    p = fmaf(p, w, 0.246640727f);
    p = fmaf(p, w, 1.50140941f);
  } else {
    w = __fsqrt_rn(w) - 3.0f;
    p = -0.000200214257f;
    p = fmaf(p, w, 0.000100950558f);
    p = fmaf(p, w, 0.00134934322f);
    p = fmaf(p, w, -0.00367342844f);
    p = fmaf(p, w, 0.00573950773f);
    p = fmaf(p, w, -0.0076224613f);
    p = fmaf(p, w, 0.00943887047f);
    p = fmaf(p, w, 1.00167406f);
    p = fmaf(p, w, 2.83297682f);
  }
  return p * x;
}

// JAX-style: bits -> [0,1) -> uniform(-1,1) -> sqrt(2)*erfinv(u)
__device__ __forceinline__ float bits_to_normal(uint32_t b) {
  float f = __uint_as_float((b >> 9) | 0x3F800000u) - 1.0f;     // [0,1)
  float u = fmaf(f, 1.99999988f, -0.99999994f);                 // (-1,1)
  return 1.41421356237f * erfinv_f(u);
}

// ---------------- Kernel A: pregenerate all noise (state-independent) ----------------
// noise layout: noise[t * 16384 + b*8 + d]  (time-major -> contiguous per-step slices)
// JAX counter mode pairs (p, p+8192): thread j produces dims for sims j and j+1024.
__global__ void lca_gen_noise(float* __restrict__ noise) {
  int gid = blockIdx.x * blockDim.x + threadIdx.x;
  if (gid >= NUM_STEPS * (NUM_SIMS / 2)) return;
  int t = gid >> 10;          // / 1024
  int j = gid & 1023;
  uint32_t k0, k1;
  // fold_in(key(42), t)
  threefry2x32(0u, 42u, (uint32_t)t, 0u, &k0, &k1);
  float* rowa = noise + (size_t)t * ROW_ELEMS + (size_t)j * NUM_DIM;   // sim j
  float* rowb = rowa + (ROW_ELEMS / 2);                                // sim j+1024
#pragma unroll
  for (int d = 0; d < NUM_DIM; d++) {
    uint32_t p = (uint32_t)(j * NUM_DIM + d);
    uint32_t b0, b1;
    threefry2x32(k0, k1, p, p + (uint32_t)(ROW_ELEMS / 2), &b0, &b1);
    rowa[d] = bits_to_normal(b0);
    rowb[d] = bits_to_normal(b1);
  }
}

// ---------------- launch ----------------
extern "C" void kernel_launch(void* const* d_in, const int* in_sizes, int n_in,
                              void* d_out, int out_size, void* d_ws, size_t ws_size,
                              hipStream_t stream) {
  (void)in_sizes; (void)n_in; (void)out_size; (void)ws_size;
  const float* input = (const float*)d_in[0];
  float* noise = (float*)d_ws;   // requires 1500*16384*4 = 98.3 MB of scratch
  float* outp  = (float*)d_out;

  const int ngen = NUM_STEPS * (NUM_SIMS / 2);   // 1,536,000 threads, 16 normals each
  lca_gen_noise<<<(ngen + 255) / 256, 256, 0, stream>>>(noise);
  lca_scan<<<NUM_SIMS / SIMS_PER_BLOCK, SIMS_PER_BLOCK, 0, stream>>>(input, noise, outp);
}